// GCNLayer_46875273069088
// MI455X (gfx1250) — compile-verified
//
#include <hip/hip_runtime.h>
#include <hip/hip_bf16.h>

typedef __attribute__((ext_vector_type(2))) float v2f;
typedef __attribute__((ext_vector_type(8))) float v8f;

#define F_DIM 64

// ---------------------------------------------------------------------------
// Kernel 0: zero the output accumulator (harness poisons d_out with 0xAA).
// ---------------------------------------------------------------------------
__global__ void zero_out_kernel(float4* __restrict__ out, int n4) {
    int i = blockIdx.x * blockDim.x + threadIdx.x;
    if (i < n4) out[i] = make_float4(0.f, 0.f, 0.f, 0.f);
}

// ---------------------------------------------------------------------------
// Kernel 1: HW = H @ W + b using V_WMMA_F32_16X16X4_F32.
// One wave computes a 16x64 strip of HW. W (64x64 f32 = 16 KB) + b staged in LDS.
// A layout (f32 16x4): lane L (L&15 = row), v0 = K = 2*(L>>4), v1 = K+1.
// B layout (f32 4x16): lane L holds col (L&15), v0 = K = 2*(L>>4), v1 = K+1.
// C/D layout: vgpr r, lane L -> row = r + 8*(L>>4), col = L&15.
// ---------------------------------------------------------------------------
__global__ void __launch_bounds__(256)
gemm_hw_kernel(const float* __restrict__ H, const float* __restrict__ W,
               const float* __restrict__ bias, float* __restrict__ HW, int M) {
    __shared__ float sW[F_DIM * F_DIM];
    __shared__ float sB[F_DIM];

    for (int i = threadIdx.x; i < F_DIM * F_DIM; i += blockDim.x) sW[i] = W[i];
    if (threadIdx.x < F_DIM) sB[threadIdx.x] = bias[threadIdx.x];
    __syncthreads();

    const int wave = threadIdx.x >> 5;          // 0..7
    const int lane = threadIdx.x & 31;
    const int tileM = (blockIdx.x * 8 + wave) * 16;
    if (tileM >= M) return;

    const int lrow = lane & 15;
    const int hi   = lane >> 4;                 // 0 or 1

    v8f acc0 = {}, acc1 = {}, acc2 = {}, acc3 = {};

    const float* hrow = H + (size_t)(tileM + lrow) * F_DIM;

#pragma unroll
    for (int k = 0; k < F_DIM; k += 4) {
        v2f a;
        a.x = hrow[k + 2 * hi];
        a.y = hrow[k + 2 * hi + 1];

        const float* w0 = &sW[(k + 2 * hi) * F_DIM];
        const float* w1 = &sW[(k + 2 * hi + 1) * F_DIM];

        v2f b0; b0.x = w0[lrow +  0]; b0.y = w1[lrow +  0];
        v2f b1; b1.x = w0[lrow + 16]; b1.y = w1[lrow + 16];
        v2f b2; b2.x = w0[lrow + 32]; b2.y = w1[lrow + 32];
        v2f b3; b3.x = w0[lrow + 48]; b3.y = w1[lrow + 48];

        acc0 = __builtin_amdgcn_wmma_f32_16x16x4_f32(false, a, false, b0, (short)0, acc0, false, false);
        acc1 = __builtin_amdgcn_wmma_f32_16x16x4_f32(false, a, false, b1, (short)0, acc1, false, false);
        acc2 = __builtin_amdgcn_wmma_f32_16x16x4_f32(false, a, false, b2, (short)0, acc2, false, false);
        acc3 = __builtin_amdgcn_wmma_f32_16x16x4_f32(false, a, false, b3, (short)0, acc3, false, false);
    }

    // Store D + bias. vgpr r, lane -> row = tileM + r + 8*hi, col = nt*16 + lrow.
#pragma unroll
    for (int r = 0; r < 8; ++r) {
        const size_t row = (size_t)(tileM + r + 8 * hi) * F_DIM;
        HW[row + lrow +  0] = acc0[r] + sB[lrow +  0];
        HW[row + lrow + 16] = acc1[r] + sB[lrow + 16];
        HW[row + lrow + 32] = acc2[r] + sB[lrow + 32];
        HW[row + lrow + 48] = acc3[r] + sB[lrow + 48];
    }
}

// ---------------------------------------------------------------------------
// Kernel 2: edge-parallel scatter. One wave32 per edge: lanes cover the 64
// features with 2 coalesced gathers from HW (L2-resident) and 2 hardware
// non-returning f32 atomic adds (global_atomic_add_f32, executed at L2).
// unsafeAtomicAdd guarantees the native fadd path (no CAS loop).
// ---------------------------------------------------------------------------
__global__ void __launch_bounds__(256)
spmm_edge_kernel(const int* __restrict__ rows, const int* __restrict__ cols,
                 const float* __restrict__ vals, const float* __restrict__ HW,
                 float* __restrict__ out, int E) {
    const int wave = threadIdx.x >> 5;
    const int lane = threadIdx.x & 31;
    const long long e = (long long)blockIdx.x * 8 + wave;
    if (e >= E) return;

    const int   r = rows[e];
    const int   c = cols[e];
    const float v = vals[e];

    const float* src = HW  + (size_t)c * F_DIM;
    float*       dst = out + (size_t)r * F_DIM;

    const float m0 = v * src[lane];
    const float m1 = v * src[lane + 32];

    unsafeAtomicAdd(dst + lane,      m0);
    unsafeAtomicAdd(dst + lane + 32, m1);
}

// ---------------------------------------------------------------------------
// Kernel 3: in-place ReLU on the accumulated output.
// ---------------------------------------------------------------------------
__global__ void relu_kernel(float4* __restrict__ out, int n4) {
    int i = blockIdx.x * blockDim.x + threadIdx.x;
    if (i < n4) {
        float4 x = out[i];
        x.x = fmaxf(x.x, 0.f);
        x.y = fmaxf(x.y, 0.f);
        x.z = fmaxf(x.z, 0.f);
        x.w = fmaxf(x.w, 0.f);
        out[i] = x;
    }
}

extern "C" void kernel_launch(void* const* d_in, const int* in_sizes, int n_in,
                              void* d_out, int out_size, void* d_ws, size_t ws_size,
                              hipStream_t stream) {
    const int*   a_rows = (const int*)d_in[0];
    const int*   a_cols = (const int*)d_in[1];
    const float* a_vals = (const float*)d_in[2];
    const float* H      = (const float*)d_in[3];
    const float* W      = (const float*)d_in[4];
    const float* b      = (const float*)d_in[5];

    const int E = in_sizes[0];
    const int M = in_sizes[3] / F_DIM;   // number of nodes

    float* HW  = (float*)d_ws;           // M x 64 f32 scratch (25.6 MB)
    float* out = (float*)d_out;

    // 0) zero the accumulator
    {
        const int n4 = out_size / 4;
        const int blocks = (n4 + 255) / 256;
        zero_out_kernel<<<blocks, 256, 0, stream>>>((float4*)out, n4);
    }
    // 1) HW = H @ W + b   (WMMA)
    {
        const int tiles  = (M + 15) / 16;
        const int blocks = (tiles + 7) / 8;     // 8 waves per block
        gemm_hw_kernel<<<blocks, 256, 0, stream>>>(H, W, b, HW, M);
    }
    // 2) scatter-add over edges (one wave per edge)
    {
        const int blocks = (E + 7) / 8;
        spmm_edge_kernel<<<blocks, 256, 0, stream>>>(a_rows, a_cols, a_vals, HW, out, E);
    }
    // 3) ReLU
    {
        const int n4 = out_size / 4;
        const int blocks = (n4 + 255) / 256;
        relu_kernel<<<blocks, 256, 0, stream>>>((float4*)out, n4);
    }
}